// ProfileLookTableEncoding_90426241450724
// MI455X (gfx1250) — compile-verified
//
#include <hip/hip_runtime.h>

typedef __attribute__((ext_vector_type(2))) float    v2f;
typedef __attribute__((ext_vector_type(8))) float    v8f;
typedef __attribute__((ext_vector_type(2))) _Float16 v2h;

#define HD 90
#define SD 30
#define VD 16
#define NLUT (HD * SD * VD)                 /* 43200 entries                    */
#define SMEM_BYTES (NLUT * 4 + NLUT * 2)    /* half2(h,s) + half(v) = 259200 B  */
#define HW (1 << 20)                        /* 1024*1024 pixels per plane       */

__device__ __forceinline__ float clamp01(float x) {
    return __builtin_fminf(__builtin_fmaxf(x, 0.f), 1.f);
}

// Branchless divide: v_rcp_f32 + one Newton-Raphson step (~1-2 ulp here; the
// denominators are in [1e-8, 1+1e-8] so no denorm/overflow corner cases).
// Avoids clang's v_cmpx/s_cbranch skip-region around precise-div sequences.
__device__ __forceinline__ float fastdiv(float a, float b) {
    float r = __builtin_amdgcn_rcpf(b);
    r = __builtin_fmaf(__builtin_fmaf(-b, r, 1.f), r, r);
    return a * r;
}

__global__ void __launch_bounds__(1024, 1)
cc_kernel(const float* __restrict__ x, const float* __restrict__ lut,
          float* __restrict__ out, int totalPix)
{
    extern __shared__ char smem[];
    v2h*      hsL = (v2h*)smem;                       // (hue_shift, sat_scale) fp16 pairs
    _Float16* vL  = (_Float16*)(smem + NLUT * 4);     // val_scale fp16

    // ---- Stage 506KB f32 LUT into 259KB of LDS as fp16 (L2-resident reads) ----
    for (int e = threadIdx.x; e < NLUT; e += blockDim.x) {
        float h = lut[e * 3 + 0];
        float s = lut[e * 3 + 1];
        float v = lut[e * 3 + 2];
        v2h p; p.x = (_Float16)h; p.y = (_Float16)s;
        hsL[e] = p;
        vL[e]  = (_Float16)v;
    }
    __syncthreads();

    const int  lane = threadIdx.x & 31;
    const int  mrow = lane & 15;
    const bool hiH  = lane >= 16;

    // A-matrix (16x4 f32) for V_WMMA_F32_16X16X4_F32:
    // lane m in [0,15]: VGPR0 = A[m][0], VGPR1 = A[m][1]; lanes 16-31: K=2,3.
    // A holds RGB2XYZ_D50 in rows 0..2, cols 0..2; zero elsewhere.
    float a0 = 0.f, a1 = 0.f;
    {
        const float M00 = 0.7976749f, M01 = 0.1351917f, M02 = 0.0313534f;
        const float M10 = 0.2880402f, M11 = 0.7118741f, M12 = 8.57e-05f;
        const float M20 = 0.f,        M21 = 0.f,        M22 = 0.82521f;
        if      (mrow == 0) { a0 = hiH ? M02 : M00; a1 = hiH ? 0.f : M01; }
        else if (mrow == 1) { a0 = hiH ? M12 : M10; a1 = hiH ? 0.f : M11; }
        else if (mrow == 2) { a0 = hiH ? M22 : M20; a1 = hiH ? 0.f : M21; }
    }
    v2f Amat; Amat.x = a0; Amat.y = a1;

    const int stride = gridDim.x * blockDim.x;
    const int tid    = blockIdx.x * blockDim.x + threadIdx.x;

    // totalPix is an exact multiple of stride here (8M / 512K = 16): every lane
    // has the same trip count -> EXEC stays all-ones around the WMMAs.
    for (int p = tid; p < totalPix; p += stride) {
        const int    bimg = p >> 20;
        const int    hw   = p & (HW - 1);
        const size_t base = (size_t)bimg * (3 * (size_t)HW) + (size_t)hw;

        float r  = clamp01(__builtin_nontemporal_load(&x[base]));
        float g  = clamp01(__builtin_nontemporal_load(&x[base + HW]));
        float bc = clamp01(__builtin_nontemporal_load(&x[base + 2 * (size_t)HW]));

        // ---- rgb2hsv: fully branch-free (selects over precomputed values) ----
        float cmax  = fmaxf(r, fmaxf(g, bc));
        float cmin  = fminf(r, fminf(g, bc));
        float delta = cmax - cmin;
        float dd    = delta + 1e-8f;
        bool  isR   = (cmax == r);
        bool  isG   = (cmax == g);
        float numR  = g - bc, numG = bc - r, numB = r - g;
        float numGB = isG ? numG : numB;
        float num   = isR ? numR : numGB;
        float q     = fastdiv(num, dd);                // |q| < 1 in the isR case
        // isR: (q % 6), q in (-1,1) -> q + (q<0 ? 6 : 0).  Else +2 / +4.
        float offR  = (q < 0.f) ? 6.f : 0.f;
        float offGB = isG ? 2.f : 4.f;
        float offs  = isR ? offR : offGB;
        float h     = (q + offs) * 60.f;
        h = (delta <= 1e-8f) ? 0.f : h;
        float sdiv = fastdiv(delta, cmax + 1e-8f);
        float s = (cmax > 1e-8f) ? sdiv : 0.f;
        float v = cmax;

        // ---- trilinear LUT lookup (all taps from LDS) ----
        float hs  = h * 0.25f;                         // * (90/360), in [0,90)
        float h0f = floorf(hs);
        float fh  = hs - h0f;
        int   i0  = (int)h0f; i0 = i0 > HD - 1 ? HD - 1 : i0;   // hs >= 0 always
        int   i1  = i0 + 1; i1 = (i1 == HD) ? 0 : i1;

        float ssc = clamp01(s) * (float)(SD - 1);
        float s0f = floorf(ssc);
        float fs  = ssc - s0f;
        int   j0  = (int)s0f; j0 = j0 > SD - 1 ? SD - 1 : j0;   // ssc >= 0 always
        int   j1  = j0 + 1;  j1 = j1 > SD - 1 ? SD - 1 : j1;

        float vsc = clamp01(v) * (float)(VD - 1);
        float v0f = floorf(vsc);
        float fv  = vsc - v0f;
        int   k0  = (int)v0f; k0 = k0 > VD - 1 ? VD - 1 : k0;   // vsc >= 0 always
        int   k1  = k0 + 1;  k1 = k1 > VD - 1 ? VD - 1 : k1;

        int e00 = i0 * (SD * VD) + j0 * VD;
        int e01 = i0 * (SD * VD) + j1 * VD;
        int e10 = i1 * (SD * VD) + j0 * VD;
        int e11 = i1 * (SD * VD) + j1 * VD;

        float w0 = 1.f - fh, w1 = fh, u0 = 1.f - fs, u1 = fs, t0 = 1.f - fv, t1 = fv;
        float acc0 = 0.f, acc1 = 0.f, acc2 = 0.f;
#define TAP(E, W) do { int e_ = (E); float w_ = (W); v2h q_ = hsL[e_];         \
        acc0 = __builtin_fmaf((float)q_.x, w_, acc0);                          \
        acc1 = __builtin_fmaf((float)q_.y, w_, acc1);                          \
        acc2 = __builtin_fmaf((float)vL[e_], w_, acc2); } while (0)
        TAP(e00 + k0, w0 * u0 * t0); TAP(e10 + k0, w1 * u0 * t0);
        TAP(e01 + k0, w0 * u1 * t0); TAP(e11 + k0, w1 * u1 * t0);
        TAP(e00 + k1, w0 * u0 * t1); TAP(e10 + k1, w1 * u0 * t1);
        TAP(e01 + k1, w0 * u1 * t1); TAP(e11 + k1, w1 * u1 * t1);
#undef TAP

        // h2 = (h + dh) mod 360 without libm fmod (cheap, branch-free)
        float hsum = h + acc0;
        float h2   = __builtin_fmaf(-360.f, floorf(hsum * (1.f / 360.f)), hsum);
        h2 = h2 < 0.f ? 0.f : (h2 >= 360.f ? h2 - 360.f : h2);
        float s2  = clamp01(s * acc1);
        float vv2 = clamp01(v * acc2);

        // ---- hsv2rgb: branch-free sector select over precomputed {c, xx, 0} ----
        float hp   = h2 * (1.f / 60.f);                // [0,6)
        float c    = vv2 * s2;
        float hpm2 = __builtin_fmaf(-2.f, floorf(hp * 0.5f), hp);   // hp mod 2
        float xx   = c * (1.f - fabsf(hpm2 - 1.f));
        float m    = vv2 - c;
        int   sec  = (int)hp; sec = sec > 5 ? 5 : sec;              // hp >= 0 always
        float rr = (sec == 0) ? c  : (sec == 1) ? xx : (sec == 2) ? 0.f
                 : (sec == 3) ? 0.f : (sec == 4) ? xx : c;
        float gg = (sec == 0) ? xx : (sec == 1) ? c  : (sec == 2) ? c
                 : (sec == 3) ? xx : 0.f;
        float bb = (sec == 0) ? 0.f : (sec == 1) ? 0.f : (sec == 2) ? xx
                 : (sec == 3) ? c  : (sec == 4) ? c  : xx;
        float R = rr + m, G = gg + m, B = bb + m;

        // ---- 3x3 color transform via V_WMMA_F32_16X16X4_F32 ----
        // Wave handles 32 pixels = two 16-pixel groups (lanes 0-15 / 16-31).
        // B-matrix (4x16): VGPR0 = rows K0(lanes0-15)=R, K2(lanes16-31)=B;
        //                  VGPR1 = rows K1=G, K3=0 (mirrors A's K half-split).
        float Rx = __shfl_xor(R, 16, 32);
        float Gx = __shfl_xor(G, 16, 32);
        float Bx = __shfl_xor(B, 16, 32);
        v2f B0, B1;
        B0.x = hiH ? Bx : R;  B0.y = hiH ? 0.f : G;   // pixels of lanes 0..15
        B1.x = hiH ? B  : Rx; B1.y = hiH ? 0.f : Gx;  // pixels of lanes 16..31
        v8f cz = {};
        v8f D0 = __builtin_amdgcn_wmma_f32_16x16x4_f32(false, Amat, false, B0,
                                                       (short)0, cz, false, false);
        v8f D1 = __builtin_amdgcn_wmma_f32_16x16x4_f32(false, Amat, false, B1,
                                                       (short)0, cz, false, false);
        // D rows 0..2 (X,Y,Z) live in D[0..2], lanes 0-15; swap group1 up.
        float x1 = __shfl_xor(D1[0], 16, 32);
        float y1 = __shfl_xor(D1[1], 16, 32);
        float z1 = __shfl_xor(D1[2], 16, 32);
        float X = hiH ? x1 : D0[0];
        float Y = hiH ? y1 : D0[1];
        float Z = hiH ? z1 : D0[2];

        __builtin_nontemporal_store(clamp01(X), &out[base]);
        __builtin_nontemporal_store(clamp01(Y), &out[base + HW]);
        __builtin_nontemporal_store(clamp01(Z), &out[base + 2 * (size_t)HW]);
    }
}

extern "C" void kernel_launch(void* const* d_in, const int* in_sizes, int n_in,
                              void* d_out, int out_size, void* d_ws, size_t ws_size,
                              hipStream_t stream) {
    (void)n_in; (void)d_ws; (void)ws_size; (void)out_size;
    const float* x   = (const float*)d_in[0];
    const float* lut = (const float*)d_in[1];
    float*       out = (float*)d_out;
    const int totalPix = in_sizes[0] / 3;   // 8 * 1024 * 1024

    // 259KB dynamic LDS per block (> default 64KB): raise the cap.
    hipFuncSetAttribute((const void*)cc_kernel,
                        hipFuncAttributeMaxDynamicSharedMemorySize, SMEM_BYTES);

    dim3 grid(512), block(1024);            // 16 pixels/thread, zero tail
    cc_kernel<<<grid, block, SMEM_BYTES, stream>>>(x, lut, out, totalPix);
}